// WSWANO_22539988369882
// MI455X (gfx1250) — compile-verified
//
#include <hip/hip_runtime.h>
#include <hip/hip_bf16.h>
#include <math.h>

// ---------------------------------------------------------------------------
// WSWA neural-operator pipeline for MI455X (gfx1250, wave32).
//   u:(8,32,256,256) f32, w:(800,32,4096) f32  ->  out:(8,32,256,256) f32
// Stages:
//   0: build 256x256 orthonormal DST-I matrix M (symmetric, involutory)
//   1: T   = U @ M[:, :80]             (f32 WMMA 16x16x4; B via M symmetry)
//   2: A80 = M[:80, :] @ T             (f32 WMMA 16x16x4)
//   3: Ypart[z] = partial pointwise    (NT 419MB weight stream, 4096 waves)
//      Y = sum_z Ypart[z]              (deterministic reduce)
//   4: S1  = Y @ M[:64, :]             (f32 WMMA 16x16x4; B via M symmetry)
//   5: out = M[:, :64] @ S1            (f32 WMMA 16x16x4; NT output store)
// Roofline: traffic ~0.55 GB (weights dominate) -> ~24 us at 23.3 TB/s;
// FLOPs ~8 G f32, trivially covered by the WMMA/VALU paths.
// ---------------------------------------------------------------------------

typedef __attribute__((ext_vector_type(2))) float v2f;
typedef __attribute__((ext_vector_type(8))) float v8f;

#define NIMG 256        // 8 batches * 32 channels
#define NN   256        // spatial size
#define TP   80         // padded truncation (68 -> 5 tiles of 16)

__device__ __forceinline__ v8f wmma4(v2f a, v2f b, v8f c) {
  // D = A(16x4,f32) * B(4x16,f32) + C(16x16,f32)
  return __builtin_amdgcn_wmma_f32_16x16x4_f32(
      /*neg_a=*/false, a, /*neg_b=*/false, b,
      /*c_mod=*/(short)0, c, /*reuse_a=*/false, /*reuse_b=*/false);
}

// -------------------------- stage 0: DST matrix ----------------------------
__global__ void build_dst(float* __restrict__ M) {
  int idx = blockIdx.x * blockDim.x + threadIdx.x;   // 0..65535
  int i = idx >> 8;
  int j = idx & 255;
  // sin(pi*(i+1)*(j+1)/257): reduce k modulo 2*257 to keep sinf accurate
  int k = ((i + 1) * (j + 1)) % 514;
  float ang = (float)k * (float)(3.14159265358979323846 / 257.0);
  M[idx] = sqrtf(2.0f / 257.0f) * sinf(ang);
}

// ------------------- stage 1: T[img] = U[img] @ M[:, :80] ------------------
// grid (5, 16, 256), block 32 (one wave per 16x16 tile)
// B fragment uses M symmetry: M[k][n] == M[n][k]  ->  contiguous b64 load.
__global__ void fwd_stage1(const float* __restrict__ u,
                           const float* __restrict__ M,
                           float* __restrict__ T) {
  const int tn   = blockIdx.x;           // m-tile (0..4)
  const int tm   = blockIdx.y;           // x-tile (0..15)
  const int img  = blockIdx.z;
  const int lane = threadIdx.x;
  const int half = lane >> 4;
  const int l16  = lane & 15;
  const float* A = u + (size_t)img * (NN * NN);
  const int arow = tm * 16 + l16;
  const int bcol = tn * 16 + l16;
  v8f acc = {0.f, 0.f, 0.f, 0.f, 0.f, 0.f, 0.f, 0.f};
  for (int k = 0; k < NN; k += 4) {
    const int ka = k + half * 2;
    v2f a = *(const v2f*)(A + (size_t)arow * NN + ka);   // 8B-aligned (ka even)
    v2f b = *(const v2f*)(M + (size_t)bcol * NN + ka);   // M symmetric
    acc = wmma4(a, b, acc);
  }
  float* C = T + (size_t)img * (NN * TP);
#pragma unroll
  for (int v = 0; v < 8; ++v) {
    const int row = tm * 16 + v + half * 8;
    C[(size_t)row * TP + bcol] = acc[v];
  }
}

// ------------------ stage 2: A80[img] = M[:80, :] @ T[img] -----------------
// grid (5, 5, 256), block 32
__global__ void fwd_stage2(const float* __restrict__ M,
                           const float* __restrict__ T,
                           float* __restrict__ A80) {
  const int tn   = blockIdx.x;           // col tile (0..4)
  const int tm   = blockIdx.y;           // row tile (0..4)
  const int img  = blockIdx.z;
  const int lane = threadIdx.x;
  const int half = lane >> 4;
  const int l16  = lane & 15;
  const float* B = T + (size_t)img * (NN * TP);
  const int arow = tm * 16 + l16;        // < 80
  const int bcol = tn * 16 + l16;        // < 80
  v8f acc = {0.f, 0.f, 0.f, 0.f, 0.f, 0.f, 0.f, 0.f};
  for (int k = 0; k < NN; k += 4) {
    const int ka = k + half * 2;
    v2f a = *(const v2f*)(M + (size_t)arow * NN + ka);
    v2f b;
    b.x = B[(size_t)ka * TP + bcol];
    b.y = B[(size_t)(ka + 1) * TP + bcol];
    acc = wmma4(a, b, acc);
  }
  float* C = A80 + (size_t)img * (TP * TP);
#pragma unroll
  for (int v = 0; v < 8; ++v) {
    const int row = tm * 16 + v + half * 8;
    C[(size_t)row * TP + bcol] = acc[v];
  }
}

// ------ stage 3a: Ypart[z][b,o,x] = sum over c-chunk z of X * W -----------
// Lanes run along x (fastest dim of W) -> the 419 MB weight stream is fully
// coalesced, each weight loaded exactly once, non-temporal so it streams
// through L2 without evicting the ~50 MB resident working set (192 MB L2).
// grid (16, 8, 4), block 256: 131072 threads = 4096 waves to keep HBM busy.
__global__ void pointwise_partial(const float* __restrict__ A80,
                                  const float* __restrict__ W,
                                  float* __restrict__ Ypart) {
  const int x  = blockIdx.x * blockDim.x + threadIdx.x;  // 0..4095
  const int o0 = blockIdx.y * 4;                         // 8 blocks of 4 outs
  const int c0 = blockIdx.z * 8;                         // 4 chunks of 8 chans
  const int xr = x >> 6;
  const int xc = x & 63;
  float acc[8][4];
#pragma unroll
  for (int b = 0; b < 8; ++b)
#pragma unroll
    for (int o = 0; o < 4; ++o) acc[b][o] = 0.f;

  for (int cc = 0; cc < 8; ++cc) {
    const int c = c0 + cc;
    for (int i5 = 0; i5 < 5; ++i5) {
#pragma unroll
      for (int j5 = 0; j5 < 5; ++j5) {
        const int soff = (xr + i5) * TP + (xc + j5);
        float xv[8];
#pragma unroll
        for (int b = 0; b < 8; ++b)
          xv[b] = A80[(size_t)(b * 32 + c) * (TP * TP) + soff];   // L2 hit
        const int i = c * 25 + i5 * 5 + j5;
        const float* wp = W + ((size_t)i * 32 + o0) * 4096 + x;
        float wv[4];
#pragma unroll
        for (int o = 0; o < 4; ++o)
          wv[o] = __builtin_nontemporal_load(wp + (size_t)o * 4096);
#pragma unroll
        for (int b = 0; b < 8; ++b)
#pragma unroll
          for (int o = 0; o < 4; ++o)
            acc[b][o] = fmaf(xv[b], wv[o], acc[b][o]);
      }
    }
  }
  float* yp = Ypart + (size_t)blockIdx.z * ((size_t)NIMG * 4096);
#pragma unroll
  for (int b = 0; b < 8; ++b)
#pragma unroll
    for (int o = 0; o < 4; ++o)
      yp[(size_t)(b * 32 + o0 + o) * 4096 + x] = acc[b][o];
}

// ------ stage 3b: Y = sum_z Ypart[z]  (deterministic, no float atomics) ----
// grid (4096), block 256
__global__ void reduce4(const float* __restrict__ Ypart, float* __restrict__ Y) {
  const size_t idx = (size_t)blockIdx.x * blockDim.x + threadIdx.x;  // < 1M
  const size_t n = (size_t)NIMG * 4096;
  Y[idx] = (Ypart[idx] + Ypart[idx + n]) + (Ypart[idx + 2 * n] + Ypart[idx + 3 * n]);
}

// ------------------ stage 4: S1[img] = Y[img] @ M[:64, :] ------------------
// grid (16, 4, 256), block 32.  B via M symmetry -> contiguous b64 loads.
__global__ void inv_stage1(const float* __restrict__ Y,
                           const float* __restrict__ M,
                           float* __restrict__ S1) {
  const int tn   = blockIdx.x;           // col tile (0..15)
  const int tm   = blockIdx.y;           // row tile (0..3)
  const int img  = blockIdx.z;
  const int lane = threadIdx.x;
  const int half = lane >> 4;
  const int l16  = lane & 15;
  const float* A = Y + (size_t)img * 4096;   // 64x64
  const int arow = tm * 16 + l16;
  const int bcol = tn * 16 + l16;
  v8f acc = {0.f, 0.f, 0.f, 0.f, 0.f, 0.f, 0.f, 0.f};
  for (int k = 0; k < 64; k += 4) {
    const int ka = k + half * 2;
    v2f a = *(const v2f*)(A + (size_t)arow * 64 + ka);
    v2f b = *(const v2f*)(M + (size_t)bcol * NN + ka);   // M symmetric
    acc = wmma4(a, b, acc);
  }
  float* C = S1 + (size_t)img * (64 * NN);
#pragma unroll
  for (int v = 0; v < 8; ++v) {
    const int row = tm * 16 + v + half * 8;
    C[(size_t)row * NN + bcol] = acc[v];
  }
}

// ----------------- stage 5: out[img] = M[:, :64] @ S1[img] -----------------
// grid (16, 16, 256), block 32.  Output written once -> non-temporal stores.
__global__ void inv_stage2(const float* __restrict__ M,
                           const float* __restrict__ S1,
                           float* __restrict__ out) {
  const int tn   = blockIdx.x;
  const int tm   = blockIdx.y;
  const int img  = blockIdx.z;
  const int lane = threadIdx.x;
  const int half = lane >> 4;
  const int l16  = lane & 15;
  const float* B = S1 + (size_t)img * (64 * NN);
  const int arow = tm * 16 + l16;
  const int bcol = tn * 16 + l16;
  v8f acc = {0.f, 0.f, 0.f, 0.f, 0.f, 0.f, 0.f, 0.f};
  for (int k = 0; k < 64; k += 4) {
    const int ka = k + half * 2;
    v2f a = *(const v2f*)(M + (size_t)arow * NN + ka);   // M[m][k], symmetric
    v2f b;
    b.x = B[(size_t)ka * NN + bcol];
    b.y = B[(size_t)(ka + 1) * NN + bcol];
    acc = wmma4(a, b, acc);
  }
  float* C = out + (size_t)img * (NN * NN);
#pragma unroll
  for (int v = 0; v < 8; ++v) {
    const int row = tm * 16 + v + half * 8;
    __builtin_nontemporal_store(acc[v], &C[(size_t)row * NN + bcol]);
  }
}

// ---------------------------------------------------------------------------
extern "C" void kernel_launch(void* const* d_in, const int* in_sizes, int n_in,
                              void* d_out, int out_size, void* d_ws, size_t ws_size,
                              hipStream_t stream) {
  const float* u = (const float*)d_in[0];   // 8*32*256*256
  const float* w = (const float*)d_in[1];   // 800*32*4096
  float* out = (float*)d_out;               // 8*32*256*256

  // Workspace layout (floats); total ~16.4M floats (~66 MB)
  float* M     = (float*)d_ws;                          // 256*256
  float* T     = M     + (size_t)NN * NN;               // 256 * 256*80
  float* A80   = T     + (size_t)NIMG * NN * TP;        // 256 * 80*80
  float* Ypart = A80   + (size_t)NIMG * TP * TP;        // 4 * 256*64*64
  float* Y     = Ypart + (size_t)4 * NIMG * 4096;       // 256 * 64*64
  float* S1    = Y     + (size_t)NIMG * 4096;           // 256 * 64*256

  build_dst<<<dim3(256), dim3(256), 0, stream>>>(M);
  fwd_stage1<<<dim3(5, 16, NIMG), dim3(32), 0, stream>>>(u, M, T);
  fwd_stage2<<<dim3(5, 5, NIMG), dim3(32), 0, stream>>>(M, T, A80);
  pointwise_partial<<<dim3(16, 8, 4), dim3(256), 0, stream>>>(A80, w, Ypart);
  reduce4<<<dim3(4096), dim3(256), 0, stream>>>(Ypart, Y);
  inv_stage1<<<dim3(16, 4, NIMG), dim3(32), 0, stream>>>(Y, M, S1);
  inv_stage2<<<dim3(16, 16, NIMG), dim3(32), 0, stream>>>(M, S1, out);
}